// GraphModel_63995012710848
// MI455X (gfx1250) — compile-verified
//
#include <hip/hip_runtime.h>
#include <hip/hip_bf16.h>
#include <math.h>

// ---------------------------------------------------------------------------
// GConvGRU (Chebyshev K=3) x2 layers + per-node linear heads, MI455X / gfx1250
//
//  - Activations node-major [N][B][F]; SpMM gathers contiguous 2KB rows/edge
//    with float4 (global_load_b128) lanes, CSR built on device (no atomics in
//    the hot loop).
//  - Gate GEMMs [80000,F]x[F,64] on V_WMMA_F32_16X16X4_F32 (fp32 matrix core).
//    Weights pre-swizzled in LDS so every B fragment is one aligned b64 load.
//    xz/xr/xh fused (shared A fragments, 12 accumulators); hz/hr fused;
//    GRU activations fused into GEMM epilogues.
// ---------------------------------------------------------------------------

#define N_NODES 10000
#define N_EDGES 160000
#define BATCH   8
#define T_WIN   16
#define FIN0    8
#define HID     64
#define ROWS    (N_NODES * BATCH)       // 80000
#define WPH     (BATCH * HID)           // 512 floats per node (h-width bufs)
#define WPX     (BATCH * FIN0)          // 64 floats per node (layer-1 x bufs)

typedef float v2f __attribute__((ext_vector_type(2)));
typedef float v8f __attribute__((ext_vector_type(8)));

// ------------------------------ small utility kernels ----------------------

__global__ void k_zero_f(float* __restrict__ p, int n) {
    int i = blockIdx.x * blockDim.x + threadIdx.x;
    if (i < n) p[i] = 0.0f;
}
__global__ void k_zero_i(int* __restrict__ p, int n) {
    int i = blockIdx.x * blockDim.x + threadIdx.x;
    if (i < n) p[i] = 0;
}

__global__ void k_deg_cnt(const int* __restrict__ rows, const float* __restrict__ w,
                          float* __restrict__ deg, int* __restrict__ cnt) {
    int e = blockIdx.x * blockDim.x + threadIdx.x;
    if (e >= N_EDGES) return;
    int r = rows[e];
    atomicAdd(&deg[r], w[e]);
    atomicAdd(&cnt[r], 1);
}

__global__ void k_lw(const int* __restrict__ rows, const int* __restrict__ cols,
                     const float* __restrict__ w, const float* __restrict__ deg,
                     float* __restrict__ lw) {
    int e = blockIdx.x * blockDim.x + threadIdx.x;
    if (e >= N_EDGES) return;
    float dr = deg[rows[e]], dc = deg[cols[e]];
    float a = (dr > 0.0f) ? rsqrtf(dr) : 0.0f;
    float b = (dc > 0.0f) ? rsqrtf(dc) : 0.0f;
    lw[e] = -a * w[e] * b;
}

__global__ void k_scan(const int* __restrict__ cnt, int* __restrict__ rowptr) {
    __shared__ int sh[1024];
    __shared__ int carry;
    const int tid = threadIdx.x;
    if (tid == 0) carry = 0;
    __syncthreads();
    for (int base = 0; base < N_NODES; base += 1024) {
        int v = (base + tid < N_NODES) ? cnt[base + tid] : 0;
        sh[tid] = v;
        __syncthreads();
        for (int off = 1; off < 1024; off <<= 1) {
            int t = (tid >= off) ? sh[tid - off] : 0;
            __syncthreads();
            sh[tid] += t;
            __syncthreads();
        }
        int c0 = carry;
        if (base + tid < N_NODES) rowptr[base + tid] = c0 + sh[tid] - v;
        int tot = sh[1023];
        __syncthreads();
        if (tid == 0) carry = c0 + tot;
        __syncthreads();
    }
    if (tid == 0) rowptr[N_NODES] = carry;
}

__global__ void k_cursor_init(int* __restrict__ cursor, const int* __restrict__ rowptr) {
    int i = blockIdx.x * blockDim.x + threadIdx.x;
    if (i < N_NODES) cursor[i] = rowptr[i];
}

__global__ void k_fill(const int* __restrict__ rows, const int* __restrict__ cols,
                       const float* __restrict__ lw, int* __restrict__ cursor,
                       int* __restrict__ colS, float* __restrict__ lwS) {
    int e = blockIdx.x * blockDim.x + threadIdx.x;
    if (e >= N_EDGES) return;
    int slot = atomicAdd(&cursor[rows[e]], 1);
    colS[slot] = cols[e];
    lwS[slot]  = lw[e];
}

// in_tensor [B,T,N,8] -> x0 [N][B][8] for one time step
__global__ void k_xload(const float* __restrict__ in, float* __restrict__ x0, int t) {
    int i = blockIdx.x * blockDim.x + threadIdx.x;   // i = n*64 + b*8 + f
    if (i >= N_NODES * WPX) return;
    int n = i >> 6, r = i & 63, b = r >> 3, f = r & 7;
    x0[i] = in[(((size_t)b * T_WIN + t) * N_NODES + n) * FIN0 + f];
}

// ------------------------------ CSR SpMM -----------------------------------
// One wave per row; lane owns WPN/32 *contiguous* channels -> float4/float2
// vector gathers (global_load_b128/b64), fully coalesced per wave.
// IMODE 0: out = P(in);  IMODE 1: out = P(in) - 0.5*src (x2 folded into GEMM)
template <int WPN, int IMODE>
__global__ __launch_bounds__(256) void k_spmm(
        const float* __restrict__ in, float* __restrict__ out,
        const float* __restrict__ src,
        const int* __restrict__ rowptr, const int* __restrict__ colS,
        const float* __restrict__ lwS) {
    constexpr int CPL = WPN / 32;          // floats per lane: 16 or 2
    const int wave = threadIdx.x >> 5;
    const int lane = threadIdx.x & 31;
    const int row  = blockIdx.x * 8 + wave;          // grid = N/8 exactly
    const size_t rb = (size_t)row * WPN + lane * CPL;
    const int s0 = rowptr[row], e0 = rowptr[row + 1];

    if constexpr (CPL == 16) {
        float4 acc[4];
#pragma unroll
        for (int i = 0; i < 4; ++i) {
            if constexpr (IMODE == 0) {
                acc[i] = make_float4(0.f, 0.f, 0.f, 0.f);
            } else {
                float4 s4 = reinterpret_cast<const float4*>(src + rb)[i];
                acc[i] = make_float4(-0.5f * s4.x, -0.5f * s4.y,
                                     -0.5f * s4.z, -0.5f * s4.w);
            }
        }
        for (int j = s0; j < e0; ++j) {
            const int   cn = colS[j];
            const float w  = lwS[j];
            const float4* ip =
                reinterpret_cast<const float4*>(in + (size_t)cn * WPN) + lane * 4;
            if (j + 1 < e0)
                __builtin_prefetch(in + (size_t)colS[j + 1] * WPN + lane * 16, 0, 3);
#pragma unroll
            for (int i = 0; i < 4; ++i) {
                float4 v = ip[i];
                acc[i].x += w * v.x; acc[i].y += w * v.y;
                acc[i].z += w * v.z; acc[i].w += w * v.w;
            }
        }
        float4* op = reinterpret_cast<float4*>(out + rb);
#pragma unroll
        for (int i = 0; i < 4; ++i) op[i] = acc[i];
    } else {
        float2 acc;
        if constexpr (IMODE == 0) {
            acc = make_float2(0.f, 0.f);
        } else {
            float2 s2 = *reinterpret_cast<const float2*>(src + rb);
            acc = make_float2(-0.5f * s2.x, -0.5f * s2.y);
        }
        for (int j = s0; j < e0; ++j) {
            const int   cn = colS[j];
            const float w  = lwS[j];
            float2 v = *(reinterpret_cast<const float2*>(in + (size_t)cn * WPN) + lane);
            if (j + 1 < e0)
                __builtin_prefetch(in + (size_t)colS[j + 1] * WPN + lane * 2, 0, 3);
            acc.x += w * v.x; acc.y += w * v.y;
        }
        *reinterpret_cast<float2*>(out + rb) = acc;
    }
}

// ------------------------------ WMMA gate GEMMs ----------------------------
// out_g[r,:64] (+)= A0@Wg[0] + A1@Wg[1] + 2*A2@Wg[2] + bias_g, fused epilogues.
// One wave computes 16x64 per gate via 4 col-tiles of V_WMMA_F32_16X16X4_F32.
// A frag (16x4 f32): lanes 0-15 -> K {kk,kk+1}, lanes 16-31 -> K {kk+2,kk+3}.
// B frag (4x16):     VGPR v, lane l -> (K = kk + v + 2*(l>=16), N = l&15).
// C/D (16x16 f32):   VGPR i, lane l -> (M = i + 8*(l>=16),      N = l&15).
// Weights pre-swizzled in LDS: (K,K+1) pair for a column stored contiguously
// -> each B fragment is a single aligned ds_load_b64 (no repack movs).

// swizzled index of element (g, s, k, col); pair (k even, k+1) contiguous
__device__ __forceinline__ int widx(int FIN, int g, int s, int k, int col) {
    return (((g * 3 + s) * (FIN >> 1) + (k >> 1)) * HID + col) * 2 + (k & 1);
}

template <int FIN, int NG>
__device__ __forceinline__ void stage_weights(float* sW,
                                              const float* const* Ws,
                                              const float* const* bs) {
    for (int g = 0; g < NG; ++g) {
        const float* Wg = Ws[g];
        for (int i = threadIdx.x; i < 3 * FIN * HID; i += 256) {
            int col = i % HID;
            int k   = (i / HID) % FIN;
            int s   = i / (HID * FIN);
            sW[widx(FIN, g, s, k, col)] = Wg[i];
        }
        for (int i = threadIdx.x; i < HID; i += 256)
            sW[NG * 3 * FIN * HID + g * HID + i] = bs[g][i];
    }
    __syncthreads();
}

template <int FIN, int NG>
__device__ __forceinline__ void gemm_core(const float* A0, const float* A1,
                                          const float* A2, const float* sW,
                                          v8f (&c)[NG][4], int rbase, int hi,
                                          int l15) {
    const float* As[3] = {A0, A1, A2};
#pragma unroll
    for (int s = 0; s < 3; ++s) {
        const float  scale = (s == 2) ? 2.0f : 1.0f;  // folds T2 = 2*P(T1)-x
        const float* A     = As[s];
        for (int kk = 0; kk < FIN; kk += 4) {
            const int kb = kk + 2 * hi;
            const float* ap = A + (size_t)(rbase + l15) * FIN + kb;
            v2f a;
            a.x = ap[0] * scale;
            a.y = ap[1] * scale;
#pragma unroll
            for (int g = 0; g < NG; ++g) {
#pragma unroll
                for (int ct = 0; ct < 4; ++ct) {
                    const int col = ct * 16 + l15;
                    v2f b = *reinterpret_cast<const v2f*>(
                        sW + widx(FIN, g, s, kb, col));
                    c[g][ct] = __builtin_amdgcn_wmma_f32_16x16x4_f32(
                        false, a, false, b, (short)0, c[g][ct], false, false);
                }
            }
        }
    }
}

// fused xz/xr/xh: three store-outputs sharing the same A fragments
template <int FIN>
__global__ __launch_bounds__(256) void k_gemm_x3(
        const float* __restrict__ A0, const float* __restrict__ A1,
        const float* __restrict__ A2,
        const float* W0, const float* b0, const float* W1, const float* b1,
        const float* W2, const float* b2,
        float* __restrict__ o0, float* __restrict__ o1, float* __restrict__ o2) {
    extern __shared__ float sW[];
    const float* Ws[3] = {W0, W1, W2};
    const float* bs[3] = {b0, b1, b2};
    stage_weights<FIN, 3>(sW, Ws, bs);
    const int wave = threadIdx.x >> 5, lane = threadIdx.x & 31;
    const int hi = lane >> 4, l15 = lane & 15;
    const int rbase = blockIdx.x * 128 + wave * 16;   // 625*128 == 80000
    v8f c[3][4];
#pragma unroll
    for (int g = 0; g < 3; ++g)
#pragma unroll
        for (int ct = 0; ct < 4; ++ct) c[g][ct] = (v8f){0, 0, 0, 0, 0, 0, 0, 0};
    gemm_core<FIN, 3>(A0, A1, A2, sW, c, rbase, hi, l15);
    float* outs[3] = {o0, o1, o2};
#pragma unroll
    for (int g = 0; g < 3; ++g) {
#pragma unroll
        for (int ct = 0; ct < 4; ++ct) {
            const int   col  = ct * 16 + l15;
            const float bias = sW[3 * 3 * FIN * HID + g * HID + col];
#pragma unroll
            for (int i = 0; i < 8; ++i) {
                const int m = rbase + i + 8 * hi;
                outs[g][(size_t)m * HID + col] = c[g][ct][i] + bias;
            }
        }
    }
}

// fused hz/hr: zb -> z = sigmoid(zb+acc+b), rb -> h*r = A0*sigmoid(rb+acc+b)
__global__ __launch_bounds__(256) void k_gemm_zr(
        const float* __restrict__ A0, const float* __restrict__ A1,
        const float* __restrict__ A2,
        const float* Wz, const float* bz, const float* Wr, const float* br,
        float* __restrict__ zb, float* __restrict__ rb) {
    extern __shared__ float sW[];
    const float* Ws[2] = {Wz, Wr};
    const float* bs[2] = {bz, br};
    stage_weights<HID, 2>(sW, Ws, bs);
    const int wave = threadIdx.x >> 5, lane = threadIdx.x & 31;
    const int hi = lane >> 4, l15 = lane & 15;
    const int rbase = blockIdx.x * 128 + wave * 16;
    v8f c[2][4];
#pragma unroll
    for (int g = 0; g < 2; ++g)
#pragma unroll
        for (int ct = 0; ct < 4; ++ct) c[g][ct] = (v8f){0, 0, 0, 0, 0, 0, 0, 0};
    gemm_core<HID, 2>(A0, A1, A2, sW, c, rbase, hi, l15);
#pragma unroll
    for (int ct = 0; ct < 4; ++ct) {
        const int   col   = ct * 16 + l15;
        const float biasz = sW[2 * 3 * HID * HID + col];
        const float biasr = sW[2 * 3 * HID * HID + HID + col];
#pragma unroll
        for (int i = 0; i < 8; ++i) {
            const int    m   = rbase + i + 8 * hi;
            const size_t idx = (size_t)m * HID + col;
            zb[idx] = 1.0f / (1.0f + expf(-(zb[idx] + c[0][ct][i] + biasz)));
            rb[idx] = A0[idx] *
                      (1.0f / (1.0f + expf(-(rb[idx] + c[1][ct][i] + biasr))));
        }
    }
}

// hh + GRU update: h = z*h + (1-z)*tanh(pb + acc + b)
__global__ __launch_bounds__(256) void k_gemm_hh(
        const float* __restrict__ A0, const float* __restrict__ A1,
        const float* __restrict__ A2,
        const float* Wh, const float* bh,
        const float* __restrict__ pb, float* __restrict__ h,
        const float* __restrict__ zb) {
    extern __shared__ float sW[];
    const float* Ws[1] = {Wh};
    const float* bs[1] = {bh};
    stage_weights<HID, 1>(sW, Ws, bs);
    const int wave = threadIdx.x >> 5, lane = threadIdx.x & 31;
    const int hi = lane >> 4, l15 = lane & 15;
    const int rbase = blockIdx.x * 128 + wave * 16;
    v8f c[1][4];
#pragma unroll
    for (int ct = 0; ct < 4; ++ct) c[0][ct] = (v8f){0, 0, 0, 0, 0, 0, 0, 0};
    gemm_core<HID, 1>(A0, A1, A2, sW, c, rbase, hi, l15);
#pragma unroll
    for (int ct = 0; ct < 4; ++ct) {
        const int   col  = ct * 16 + l15;
        const float bias = sW[3 * HID * HID + col];
#pragma unroll
        for (int i = 0; i < 8; ++i) {
            const int    m   = rbase + i + 8 * hi;
            const size_t idx = (size_t)m * HID + col;
            float pre = pb[idx] + c[0][ct][i] + bias;
            float zz  = zb[idx];
            float hv  = h[idx];
            h[idx] = zz * hv + (1.0f - zz) * tanhf(pre);
        }
    }
}

// ------------------------------ output heads -------------------------------

__global__ void k_heads(const float* __restrict__ h2,
                        const float* __restrict__ muw, const float* __restrict__ mub,
                        const float* __restrict__ sgw, const float* __restrict__ sgb,
                        float* __restrict__ out) {
    int i = blockIdx.x * blockDim.x + threadIdx.x;
    if (i >= BATCH * N_NODES * 2) return;
    const int o = i & 1;
    const int n = (i >> 1) % N_NODES;
    const int b = i / (2 * N_NODES);
    const float* hv = h2 + ((size_t)n * BATCH + b) * HID;
    float sm = mub[n * 2 + o], ss = sgb[n * 2 + o];
    for (int f = 0; f < HID; ++f) {
        float h = hv[f];
        sm += h * muw[((size_t)n * HID + f) * 2 + o];
        ss += h * sgw[((size_t)n * HID + f) * 2 + o];
    }
    out[((size_t)b * N_NODES + n) * 2 + o] = sm;
    float sp = (ss > 20.0f) ? ss : log1pf(expf(ss));
    out[(size_t)BATCH * N_NODES * 2 + ((size_t)b * N_NODES + n) * 2 + o] = sp;
}

__global__ void k_pool(const float* __restrict__ h2, float* __restrict__ pooled) {
    int tid = threadIdx.x;                 // tid = b*64 + f, 512 threads
    float acc = 0.0f;
    for (int n = 0; n < N_NODES; ++n) acc += h2[(size_t)n * WPH + tid];
    pooled[tid] = acc / (float)N_NODES;
}

__global__ void k_softmax(const float* __restrict__ pooled, float* __restrict__ out) {
    __shared__ float sh[512];
    __shared__ float mx[BATCH], sm[BATCH];
    int tid = threadIdx.x;                 // 512 threads
    int b = tid >> 6, f = tid & 63;
    float v = pooled[tid];
    sh[tid] = v;
    __syncthreads();
    if (f == 0) {
        float m = -1e30f;
        for (int j = 0; j < 64; ++j) m = fmaxf(m, sh[b * 64 + j]);
        mx[b] = m;
    }
    __syncthreads();
    float e = expf(v - mx[b]);
    sh[tid] = e;
    __syncthreads();
    if (f == 0) {
        float s = 0.0f;
        for (int j = 0; j < 64; ++j) s += sh[b * 64 + j];
        sm[b] = s;
    }
    __syncthreads();
    out[(size_t)2 * BATCH * N_NODES * 2 + tid] = e / sm[b];
}

// ------------------------------ host side ----------------------------------

struct Gate { const float* W; const float* b; };

// dynamic-LDS byte sizes
static inline size_t shmem_ng(int fin, int ng) {
    return (size_t)(ng * 3 * fin * HID + ng * HID) * sizeof(float);
}

// gates in order: xz, hz, xr, hr, xh, hh
static void run_layer(hipStream_t st, bool fin8,
                      const float* x, const float* T1x, const float* T2x,
                      float* h, float* T1, float* T2,
                      float* zb, float* rb, float* pb, const Gate* g,
                      const int* rowptr, const int* colS, const float* lwS) {
    k_spmm<WPH, 0><<<N_NODES / 8, 256, 0, st>>>(h, T1, nullptr, rowptr, colS, lwS);
    k_spmm<WPH, 1><<<N_NODES / 8, 256, 0, st>>>(T1, T2, h, rowptr, colS, lwS);
    if (fin8)
        k_gemm_x3<FIN0><<<625, 256, shmem_ng(FIN0, 3), st>>>(
            x, T1x, T2x, g[0].W, g[0].b, g[2].W, g[2].b, g[4].W, g[4].b,
            zb, rb, pb);
    else
        k_gemm_x3<HID><<<625, 256, shmem_ng(HID, 3), st>>>(
            x, T1x, T2x, g[0].W, g[0].b, g[2].W, g[2].b, g[4].W, g[4].b,
            zb, rb, pb);
    k_gemm_zr<<<625, 256, shmem_ng(HID, 2), st>>>(
        h, T1, T2, g[1].W, g[1].b, g[3].W, g[3].b, zb, rb);
    k_spmm<WPH, 0><<<N_NODES / 8, 256, 0, st>>>(rb, T1, nullptr, rowptr, colS, lwS);
    k_spmm<WPH, 1><<<N_NODES / 8, 256, 0, st>>>(T1, T2, rb, rowptr, colS, lwS);
    k_gemm_hh<<<625, 256, shmem_ng(HID, 1), st>>>(
        rb, T1, T2, g[5].W, g[5].b, pb, h, zb);
}

extern "C" void kernel_launch(void* const* d_in, const int* in_sizes, int n_in,
                              void* d_out, int out_size, void* d_ws, size_t ws_size,
                              hipStream_t stream) {
    (void)in_sizes; (void)n_in; (void)out_size; (void)ws_size;

    // ---- inputs (top-level in file order; params flattened as JAX pytree:
    //      sorted dict keys; gates hh,hr,hz,xh,xr,xz; each gate W then b) ----
    const float* in_tensor = (const float*)d_in[0];   // [B,T,N,8]
    const int*   e_rows    = (const int*)d_in[1];     // edge_index[0]
    const int*   e_cols    = ((const int*)d_in[1]) + N_EDGES;
    const float* e_w       = (const float*)d_in[2];   // [E]
    auto L = [&](int i) { return (const float*)d_in[i]; };
    // layer0 leaves 3..14, layer1 leaves 15..26 (hh.W, hh.b, hr.W, ... xz.b)
    Gate g0[6] = { {L(13), L(14)}, {L(7),  L(8)},  {L(11), L(12)},
                   {L(5),  L(6)},  {L(9),  L(10)}, {L(3),  L(4)} };   // xz,hz,xr,hr,xh,hh
    Gate g1[6] = { {L(25), L(26)}, {L(19), L(20)}, {L(23), L(24)},
                   {L(17), L(18)}, {L(21), L(22)}, {L(15), L(16)} };
    const float* mu_b  = L(27);   // [N,2]
    const float* mu_w  = L(28);   // [N,64,2]
    const float* sig_b = L(29);
    const float* sig_w = L(30);

    float* out = (float*)d_out;   // [mu | sigma | mix] = 160000+160000+512

    // ---- workspace carve-up ----
    unsigned char* wsb = (unsigned char*)d_ws;
    size_t off = 0;
    auto alloc = [&](size_t bytes) -> void* {
        void* p = wsb + off;
        off = (off + bytes + 255) & ~(size_t)255;
        return p;
    };
    float* deg    = (float*)alloc(sizeof(float) * N_NODES);
    float* lw     = (float*)alloc(sizeof(float) * N_EDGES);
    int*   rowptr = (int*)alloc(sizeof(int) * (N_NODES + 1));
    int*   cursor = (int*)alloc(sizeof(int) * N_NODES);      // also used as cnt
    int*   colS   = (int*)alloc(sizeof(int) * N_EDGES);
    float* lwS    = (float*)alloc(sizeof(float) * N_EDGES);
    float* x0     = (float*)alloc(sizeof(float) * N_NODES * WPX);
    float* T1x0   = (float*)alloc(sizeof(float) * N_NODES * WPX);
    float* T2x0   = (float*)alloc(sizeof(float) * N_NODES * WPX);
    float* h1     = (float*)alloc(sizeof(float) * (size_t)N_NODES * WPH);
    float* h2     = (float*)alloc(sizeof(float) * (size_t)N_NODES * WPH);
    float* T1     = (float*)alloc(sizeof(float) * (size_t)N_NODES * WPH);
    float* T2     = (float*)alloc(sizeof(float) * (size_t)N_NODES * WPH);
    float* T1x2   = (float*)alloc(sizeof(float) * (size_t)N_NODES * WPH);
    float* T2x2   = (float*)alloc(sizeof(float) * (size_t)N_NODES * WPH);
    float* zb     = (float*)alloc(sizeof(float) * (size_t)N_NODES * WPH);
    float* rb     = (float*)alloc(sizeof(float) * (size_t)N_NODES * WPH);
    float* pb     = (float*)alloc(sizeof(float) * (size_t)N_NODES * WPH);
    float* pooled = (float*)alloc(sizeof(float) * 512);

    const int EB = (N_EDGES + 255) / 256;   // 625
    const int NB = (N_NODES + 255) / 256;   // 40

    // ---- graph preprocessing: scaled laplacian weights + CSR ----
    k_zero_f<<<NB, 256, 0, stream>>>(deg, N_NODES);
    k_zero_i<<<NB, 256, 0, stream>>>(cursor, N_NODES);                 // cnt
    k_deg_cnt<<<EB, 256, 0, stream>>>(e_rows, e_w, deg, cursor);
    k_lw<<<EB, 256, 0, stream>>>(e_rows, e_cols, e_w, deg, lw);
    k_scan<<<1, 1024, 0, stream>>>(cursor, rowptr);
    k_cursor_init<<<NB, 256, 0, stream>>>(cursor, rowptr);
    k_fill<<<EB, 256, 0, stream>>>(e_rows, e_cols, lw, cursor, colS, lwS);

    // ---- init hidden states ----
    const int HB = ((int)((size_t)N_NODES * WPH) + 255) / 256;
    k_zero_f<<<HB, 256, 0, stream>>>(h1, N_NODES * WPH);
    k_zero_f<<<HB, 256, 0, stream>>>(h2, N_NODES * WPH);

    // ---- time recurrence ----
    for (int t = 0; t < T_WIN; ++t) {
        // layer-1 input + its Chebyshev terms (width 64 = B*8)
        k_xload<<<(N_NODES * WPX + 255) / 256, 256, 0, stream>>>(in_tensor, x0, t);
        k_spmm<WPX, 0><<<N_NODES / 8, 256, 0, stream>>>(x0, T1x0, nullptr, rowptr, colS, lwS);
        k_spmm<WPX, 1><<<N_NODES / 8, 256, 0, stream>>>(T1x0, T2x0, x0, rowptr, colS, lwS);
        run_layer(stream, true, x0, T1x0, T2x0, h1, T1, T2, zb, rb, pb, g0,
                  rowptr, colS, lwS);
        // layer-2 input = h1; its Chebyshev terms (width 512)
        k_spmm<WPH, 0><<<N_NODES / 8, 256, 0, stream>>>(h1, T1x2, nullptr, rowptr, colS, lwS);
        k_spmm<WPH, 1><<<N_NODES / 8, 256, 0, stream>>>(T1x2, T2x2, h1, rowptr, colS, lwS);
        run_layer(stream, false, h1, T1x2, T2x2, h2, T1, T2, zb, rb, pb, g1,
                  rowptr, colS, lwS);
    }

    // ---- heads: mu, softplus(sigma), softmax(mean-pool) ----
    k_heads<<<(BATCH * N_NODES * 2 + 255) / 256, 256, 0, stream>>>(
        h2, mu_w, mu_b, sig_w, sig_b, out);
    k_pool<<<1, 512, 0, stream>>>(h2, pooled);
    k_softmax<<<1, 512, 0, stream>>>(pooled, out);
}